// Router_64003602645350
// MI455X (gfx1250) — compile-verified
//
// MI455X (gfx1250) fused hex-router kernel.
// Memory-bound: streams ~880MB of per-edge/per-node weights once per call ->
// ~38us floor @ 23.3TB/s HBM. Stream is 4.5x the 192MB L2 with zero reuse, so
// all weight traffic uses non-temporal (TH=NT) loads; the small reused arrays
// (H, coords, betas) stay on the default RT policy.
// One workgroup per destination node (edges contiguous: dst = e/6), 7 wave32s:
//   waves 0..5: edge matvecs msg_e = W_edge[e]@H[src], K_e = key_W[e]@H[src]
//   wave 6:     q_r = query_W[r]@H[r]
// Matvecs use V_WMMA_F32_16X16X4_F32 (CDNA5 fp32 WMMA); B = h-chunk broadcast
// across all 16 N-columns. Compute sits far under the bandwidth roofline.

#include <hip/hip_runtime.h>
#include <math.h>

#define DIM 64
#define DEG 6

typedef float v2f __attribute__((ext_vector_type(2)));
typedef float v8f __attribute__((ext_vector_type(8)));

#if defined(__AMDGCN__) && __has_builtin(__builtin_amdgcn_wmma_f32_16x16x4_f32)
#define USE_WMMA_F32 1
#endif

__device__ __forceinline__ float wave_sum32(float v) {
    v += __shfl_xor(v, 16);
    v += __shfl_xor(v, 8);
    v += __shfl_xor(v, 4);
    v += __shfl_xor(v, 2);
    v += __shfl_xor(v, 1);
    return v;
}

__device__ __forceinline__ float softplusf(float x) {
    return (x > 20.0f) ? x : log1pf(expf(x));
}

// Streaming (non-temporal) v2f load: weights are touched exactly once.
__device__ __forceinline__ v2f nt_load2(const float* p) {
    return __builtin_nontemporal_load((const v2f*)p);
}

// out[0..63] (LDS) = scale * (Wm(64x64 row-major, global) @ h(64, LDS))
// Entire wave32 participates; must be called with full EXEC (wave-uniform branch).
__device__ __forceinline__ void wave_matvec64(const float* __restrict__ Wm,
                                              const float* __restrict__ h,
                                              float* __restrict__ out,
                                              float scale, int lane) {
#ifdef USE_WMMA_F32
    const int m  = lane & 15;
    const int hi = lane >> 4;            // 0: lanes 0-15 (K=0,1) 1: lanes 16-31 (K=2,3)
    const int c  = hi ? 2 : 0;
    v8f acc0 = {}, acc1 = {}, acc2 = {}, acc3 = {};
#pragma unroll
    for (int kb = 0; kb < 16; ++kb) {
        const int col = kb * 4 + c;
        v2f b;                            // B[k,n] = h[k] for all n (broadcast columns)
        b.x = h[col];
        b.y = h[col + 1];
        v2f a0 = nt_load2(Wm + (size_t)(m     ) * DIM + col);
        v2f a1 = nt_load2(Wm + (size_t)(m + 16) * DIM + col);
        v2f a2 = nt_load2(Wm + (size_t)(m + 32) * DIM + col);
        v2f a3 = nt_load2(Wm + (size_t)(m + 48) * DIM + col);
        acc0 = __builtin_amdgcn_wmma_f32_16x16x4_f32(false, a0, false, b, (short)0, acc0, false, false);
        acc1 = __builtin_amdgcn_wmma_f32_16x16x4_f32(false, a1, false, b, (short)0, acc1, false, false);
        acc2 = __builtin_amdgcn_wmma_f32_16x16x4_f32(false, a2, false, b, (short)0, acc2, false, false);
        acc3 = __builtin_amdgcn_wmma_f32_16x16x4_f32(false, a3, false, b, (short)0, acc3, false, false);
    }
    // D layout (16x16 f32): VGPR v, lanes 0-15 -> row v, lanes 16-31 -> row v+8.
    // All 16 columns identical -> lanes 0-7 / 16-23 write one element each.
    if (m < 8) {
        const int base = hi ? 8 : 0;
        out[ 0 + base + m] = scale * acc0[m];
        out[16 + base + m] = scale * acc1[m];
        out[32 + base + m] = scale * acc2[m];
        out[48 + base + m] = scale * acc3[m];
    }
#else
    // Fallback (also used by the host compile pass): plain FMA matvec.
    float s0 = 0.0f, s1 = 0.0f;
    for (int j = 0; j < DIM; ++j) {
        const float hv = h[j];
        s0 = fmaf(Wm[(size_t)lane * DIM + j], hv, s0);
        s1 = fmaf(Wm[(size_t)(lane + 32) * DIM + j], hv, s1);
    }
    out[lane]      = scale * s0;
    out[lane + 32] = scale * s1;
#endif
}

__global__ __launch_bounds__(224) void hex_router_kernel(
        const float* __restrict__ H,            // [R,64]
        const unsigned char* __restrict__ mask, // [R] bool
        const float* __restrict__ coords,       // [R,2]
        const float* __restrict__ W_edge,       // [E,64,64]
        const float* __restrict__ key_W,        // [E,64,64]
        const float* __restrict__ rawP,         // [E,64]
        const float* __restrict__ query_W,      // [R,64,64]
        const float* __restrict__ W_reg,        // [8,2]
        const float* __restrict__ beta_cos,     // [8]
        const float* __restrict__ beta_sin,     // [8]
        const int* __restrict__ src_idx,        // [E]
        float* __restrict__ out)                // [R,64]
{
    __shared__ float sh_hsrc[DEG][DIM];
    __shared__ float sh_hdst[DIM];
    __shared__ float sh_q[DIM];
    __shared__ float sh_msg[DEG][DIM];
    __shared__ float sh_k[DEG][DIM];
    __shared__ float sh_logw[DEG];

    const int r    = blockIdx.x;
    const int wave = threadIdx.x >> 5;   // 0..6
    const int lane = threadIdx.x & 31;

    // ---- Phase 0: stage H vectors into LDS ----
    int e = 0, s = 0;
    bool em = false;
    if (wave < DEG) {
        e = r * DEG + wave;
        s = src_idx[e];
        sh_hsrc[wave][lane]      = H[(size_t)s * DIM + lane];
        sh_hsrc[wave][lane + 32] = H[(size_t)s * DIM + lane + 32];
        em = (mask[s] != 0);
    } else {
        sh_hdst[lane]      = H[(size_t)r * DIM + lane];
        sh_hdst[lane + 32] = H[(size_t)r * DIM + lane + 32];
    }
    __syncthreads();

    // ---- Fourier multiplicative bias (per edge, cheap) ----
    float factor = 1.0f;
    if (wave < DEG) {
        const float dx = coords[2 * r]     - coords[2 * s];
        const float dy = coords[2 * r + 1] - coords[2 * s + 1];
        float t = 0.0f;
        if (lane < 8) {
            const float S = dx * W_reg[2 * lane] + dy * W_reg[2 * lane + 1];
            t = cosf(S) * beta_cos[lane] + sinf(S) * beta_sin[lane];
        }
        const float b = wave_sum32(t) * 0.35355339059327373f;  // 1/sqrt(8)
        factor = 1.0f + 0.1f * b;
    }

    // ---- Phase A: msg matvecs (waves 0-5), q matvec (wave 6) ----
    if (wave < DEG) {
        wave_matvec64(W_edge + (size_t)e * (DIM * DIM), sh_hsrc[wave], sh_msg[wave], factor, lane);
    } else {
        wave_matvec64(query_W + (size_t)r * (DIM * DIM), sh_hdst, sh_q, 1.0f, lane);
    }
    __syncthreads();

    // ---- Phase B: key matvecs ----
    if (wave < DEG) {
        wave_matvec64(key_W + (size_t)e * (DIM * DIM), sh_hsrc[wave], sh_k[wave], 1.0f, lane);
    }

    // ---- Phase C: per-edge score / Mahalanobis / logw ----
    if (wave < DEG) {
        const int d0 = lane, d1 = lane + 32;
        const float kv0 = sh_k[wave][d0], kv1 = sh_k[wave][d1];
        const float score = wave_sum32(sh_q[d0] * kv0 + sh_q[d1] * kv1) * 0.125f; // /sqrt(64)
        const float m0 = sh_msg[wave][d0], m1 = sh_msg[wave][d1];
        const float r0 = m0 - sh_hdst[d0], r1 = m1 - sh_hdst[d1];
        // raw_P_edge is a one-shot 6MB stream as well -> NT loads.
        const float p0 = softplusf(__builtin_nontemporal_load(rawP + (size_t)e * DIM + d0));
        const float p1 = softplusf(__builtin_nontemporal_load(rawP + (size_t)e * DIM + d1));
        float mah = wave_sum32(r0 * r0 * p0 + r1 * r1 * p1);
        mah = fminf(mah, 160.0f);                       // MAX_EXP / 0.5
        float logw = score - 0.5f * mah;
        if (!em) logw = -INFINITY;
        if (lane == 0) sh_logw[wave] = logw;
    }
    __syncthreads();

    // ---- Phase D: 6-way segment softmax + weighted sum ----
    const int tid = threadIdx.x;
    if (tid < DIM) {
        float lw[DEG];
        float mx = -INFINITY;
#pragma unroll
        for (int w = 0; w < DEG; ++w) {
            lw[w] = sh_logw[w];
            mx = fmaxf(mx, lw[w]);
        }
        float acc = 0.0f;
        if (mx != -INFINITY) {
            float ex[DEG];
            float sum = 0.0f;
#pragma unroll
            for (int w = 0; w < DEG; ++w) {
                ex[w] = (lw[w] == -INFINITY) ? 0.0f : expf(lw[w] - mx);
                sum += ex[w];
            }
            const float inv = 1.0f / fmaxf(sum, 1e-12f);
#pragma unroll
            for (int w = 0; w < DEG; ++w) {
                acc += (ex[w] * inv) * sh_msg[w][tid];
            }
        }
        // Write-once output: non-temporal store.
        __builtin_nontemporal_store(acc, out + (size_t)r * DIM + tid);
    }
}

extern "C" void kernel_launch(void* const* d_in, const int* in_sizes, int n_in,
                              void* d_out, int out_size, void* d_ws, size_t ws_size,
                              hipStream_t stream) {
    const float*         H        = (const float*)d_in[0];
    const unsigned char* maskp    = (const unsigned char*)d_in[1];
    const float*         coords   = (const float*)d_in[2];
    const float*         W_edge   = (const float*)d_in[3];
    const float*         key_W    = (const float*)d_in[4];
    const float*         rawP     = (const float*)d_in[5];
    const float*         query_W  = (const float*)d_in[6];
    const float*         W_reg    = (const float*)d_in[7];
    const float*         beta_cos = (const float*)d_in[8];
    const float*         beta_sin = (const float*)d_in[9];
    const int*           src_idx  = (const int*)d_in[10];
    // d_in[11] = dst_idx: unused (dst = e / DEG by construction)

    const int R = in_sizes[0] / DIM;     // 4096
    hex_router_kernel<<<dim3(R), dim3(7 * 32), 0, stream>>>(
        H, maskp, coords, W_edge, key_W, rawP, query_W,
        W_reg, beta_cos, beta_sin, src_idx, (float*)d_out);
    (void)d_ws; (void)ws_size; (void)n_in; (void)out_size;
}